// ModularSDIModel_68255620268131
// MI455X (gfx1250) — compile-verified
//
#include <hip/hip_runtime.h>
#include <hip/hip_bf16.h>

// Problem constants (match reference)
#define B_  256
#define N_  2048
#define D_  128
#define P_  2048
#define C_  10
#define LN_EPS   1e-5f
#define NORM_EPS 1e-12f

typedef __attribute__((ext_vector_type(2))) float v2f;
typedef __attribute__((ext_vector_type(8))) float v8f;

// ---------------------------------------------------------------------------
// Kernel 1: fused  h = X@W1+b1  ->  LayerNorm+ReLU  ->  pair gather ->
//           normalize(za-zb), normalize(za+zb) -> mean  ==> sdi (B, D)
// One workgroup (256 threads = 8 wave32) per batch b.
// X[b] (16KB), pairs (16KB), per-row LN stats (16KB) all live in LDS, so z is
// never materialized in HBM (saves ~0.5 GB of DRAM round-trip).
// ---------------------------------------------------------------------------
__global__ __launch_bounds__(256) void sdi_pair_kernel(
    const float* __restrict__ X,      // (B, N, 2)
    const int*   __restrict__ pairs,  // (P, 2)
    const float* __restrict__ W1,     // (2, D)
    const float* __restrict__ b1,     // (D)
    const float* __restrict__ gamma,  // (D)
    const float* __restrict__ beta,   // (D)
    float*       __restrict__ sdi_out)// (B, D)
{
    __shared__ float sX[N_ * 2];        // 16 KB
    __shared__ int   sPairs[P_ * 2];    // 16 KB
    __shared__ float sMu[N_];           // 8 KB
    __shared__ float sRstd[N_];         // 8 KB
    __shared__ float sAcc[D_];          // 512 B

    const int b    = blockIdx.x;
    const int tid  = threadIdx.x;
    const int lane = tid & 31;
    const int wave = tid >> 5;          // 0..7
    const int d0   = lane * 4;          // each lane owns 4 contiguous channels

    // Cooperative stage of X[b] and pairs into LDS (vectorized 16B).
    const float* Xb = X + (size_t)b * (N_ * 2);
    for (int i = tid; i < (N_ * 2) / 4; i += 256)
        ((float4*)sX)[i] = ((const float4*)Xb)[i];
    for (int i = tid; i < (P_ * 2) / 4; i += 256)
        ((int4*)sPairs)[i] = ((const int4*)pairs)[i];
    if (tid < D_) sAcc[tid] = 0.f;

    // Per-lane constants for its 4 channels.
    float w0[4], w1v[4], b1v[4], gv[4], bv[4];
    #pragma unroll
    for (int j = 0; j < 4; ++j) {
        w0[j]  = W1[d0 + j];
        w1v[j] = W1[D_ + d0 + j];
        b1v[j] = b1[d0 + j];
        gv[j]  = gamma[d0 + j];
        bv[j]  = beta[d0 + j];
    }
    __syncthreads();

    // Pass 1: LN statistics for every row n (each wave takes n = wave + 8k).
    for (int n = wave; n < N_; n += 8) {
        float x0 = sX[2 * n], x1 = sX[2 * n + 1];
        float s = 0.f, s2 = 0.f;
        #pragma unroll
        for (int j = 0; j < 4; ++j) {
            float h = fmaf(x0, w0[j], fmaf(x1, w1v[j], b1v[j]));
            s += h;
            s2 = fmaf(h, h, s2);
        }
        #pragma unroll
        for (int off = 16; off; off >>= 1) {
            s  += __shfl_xor(s,  off, 32);
            s2 += __shfl_xor(s2, off, 32);
        }
        float mu  = s * (1.f / 128.f);
        float var = fmaf(-mu, mu, s2 * (1.f / 128.f));
        if (lane == 0) {
            sMu[n]   = mu;
            sRstd[n] = rsqrtf(var + LN_EPS);
        }
    }
    __syncthreads();

    // Pass 2: pairs. One wave32 per pair; rows recomputed from LDS-resident X.
    float acc[4] = {0.f, 0.f, 0.f, 0.f};
    for (int p = wave; p < P_; p += 8) {
        int ia = sPairs[2 * p], ib = sPairs[2 * p + 1];
        float xa0 = sX[2 * ia], xa1 = sX[2 * ia + 1];
        float xb0 = sX[2 * ib], xb1 = sX[2 * ib + 1];
        float mua = sMu[ia], ra = sRstd[ia];
        float mub = sMu[ib], rb = sRstd[ib];

        float nd = 0.f, ns = 0.f;
        float diff[4], ssum[4];
        #pragma unroll
        for (int j = 0; j < 4; ++j) {
            float ha = fmaf(xa0, w0[j], fmaf(xa1, w1v[j], b1v[j]));
            float hb = fmaf(xb0, w0[j], fmaf(xb1, w1v[j], b1v[j]));
            float za = fmaxf(fmaf(gv[j], (ha - mua) * ra, bv[j]), 0.f);
            float zb = fmaxf(fmaf(gv[j], (hb - mub) * rb, bv[j]), 0.f);
            diff[j] = za - zb;
            ssum[j] = za + zb;
            nd = fmaf(diff[j], diff[j], nd);
            ns = fmaf(ssum[j], ssum[j], ns);
        }
        #pragma unroll
        for (int off = 16; off; off >>= 1) {
            nd += __shfl_xor(nd, off, 32);
            ns += __shfl_xor(ns, off, 32);
        }
        float rnd = 1.f / fmaxf(sqrtf(nd), NORM_EPS);
        float rns = 1.f / fmaxf(sqrtf(ns), NORM_EPS);
        #pragma unroll
        for (int j = 0; j < 4; ++j)
            acc[j] = fmaf(diff[j], rnd, fmaf(ssum[j], rns, acc[j]));
    }

    // Combine 8 waves via LDS float atomics (ds_add_f32), then mean over 2P.
    #pragma unroll
    for (int j = 0; j < 4; ++j)
        atomicAdd(&sAcc[d0 + j], acc[j]);
    __syncthreads();
    if (tid < D_)
        sdi_out[(size_t)b * D_ + tid] = sAcc[tid] * (1.f / (2.f * P_));
}

// ---------------------------------------------------------------------------
// Kernel 2: MLP head with V_WMMA_F32_16X16X4_F32.
//   hidden = relu(sdi @ Wm1 + bm1)            (256 x 128)
//   logits = sdi @ mem^T + hidden @ Wm2 + bm2 (256 x 10, N padded to 16)
// One wave32 per 16-row M tile (grid = 16). EXEC kept all-ones around WMMA;
// N-padding handled with branchless selects.
//
// f32 16x16x4 fragment layouts (ISA 7.12.2):
//   A (16x4):  lane<16 -> {A[M=l][k0], A[M=l][k0+1]}, lane>=16 -> K=k0+2,k0+3
//   B (4x16):  v0 = B[k0+2*half][n0+l], v1 = B[k0+2*half+1][n0+l]
//   C/D(16x16): vgpr r = C[M = 8*half + r][N = n0 + l]
// ---------------------------------------------------------------------------
__global__ __launch_bounds__(32) void sdi_head_kernel(
    const float* __restrict__ sdi,   // (B, D)
    const float* __restrict__ Wm1,   // (D, D)
    const float* __restrict__ bm1,   // (D)
    const float* __restrict__ Wm2,   // (D, C)
    const float* __restrict__ bm2,   // (C)
    const float* __restrict__ mem,   // (C, D)
    float*       __restrict__ logits)// (B, C)
{
    __shared__ float hid[16 * D_];   // 8 KB hidden tile

    const int lane = threadIdx.x;    // 0..31
    const int half = lane >> 4;      // 0 / 1
    const int l    = lane & 15;
    const int m0   = blockIdx.x * 16;

    const float* sdiRow = sdi + (size_t)(m0 + l) * D_;

    // ---- hidden = relu(sdi_tile @ Wm1 + bm1), 8 N-tiles of 16 ----
    for (int n0 = 0; n0 < D_; n0 += 16) {
        v8f accv = {};
        #pragma unroll 4
        for (int k0 = 0; k0 < D_; k0 += 4) {
            const int kA = k0 + 2 * half;
            v2f a, bf;
            a.x  = sdiRow[kA];
            a.y  = sdiRow[kA + 1];
            bf.x = Wm1[kA * D_ + n0 + l];
            bf.y = Wm1[(kA + 1) * D_ + n0 + l];
            accv = __builtin_amdgcn_wmma_f32_16x16x4_f32(
                false, a, false, bf, (short)0, accv, false, false);
        }
        float bias = bm1[n0 + l];
        #pragma unroll
        for (int r = 0; r < 8; ++r) {
            int row = half * 8 + r;
            hid[row * D_ + n0 + l] = fmaxf(accv[r] + bias, 0.f);
        }
    }
    __syncthreads();

    // ---- logits = sdi_tile @ mem^T + hidden_tile @ Wm2 (N padded 10->16) ----
    const int n     = l;             // output column, valid when < C_
    const bool nval = (n < C_);
    v8f accv = {};
    #pragma unroll 4
    for (int k0 = 0; k0 < D_; k0 += 4) {     // sdi @ mem^T
        const int kA = k0 + 2 * half;
        v2f a, bf;
        a.x  = sdiRow[kA];
        a.y  = sdiRow[kA + 1];
        bf.x = nval ? mem[n * D_ + kA]     : 0.f;
        bf.y = nval ? mem[n * D_ + kA + 1] : 0.f;
        accv = __builtin_amdgcn_wmma_f32_16x16x4_f32(
            false, a, false, bf, (short)0, accv, false, false);
    }
    #pragma unroll 4
    for (int k0 = 0; k0 < D_; k0 += 4) {     // hidden @ Wm2
        const int kA = k0 + 2 * half;
        v2f a, bf;
        a.x  = hid[l * D_ + kA];
        a.y  = hid[l * D_ + kA + 1];
        bf.x = nval ? Wm2[kA * C_ + n]       : 0.f;
        bf.y = nval ? Wm2[(kA + 1) * C_ + n] : 0.f;
        accv = __builtin_amdgcn_wmma_f32_16x16x4_f32(
            false, a, false, bf, (short)0, accv, false, false);
    }

    if (nval) {
        float bias = bm2[n];
        #pragma unroll
        for (int r = 0; r < 8; ++r) {
            int row = m0 + half * 8 + r;
            logits[row * C_ + n] = accv[r] + bias;
        }
    }
}

// ---------------------------------------------------------------------------
extern "C" void kernel_launch(void* const* d_in, const int* in_sizes, int n_in,
                              void* d_out, int out_size, void* d_ws, size_t ws_size,
                              hipStream_t stream) {
    const float* X     = (const float*)d_in[0];
    const int*   pairs = (const int*)  d_in[1];
    const float* W1    = (const float*)d_in[2];
    const float* b1    = (const float*)d_in[3];
    const float* gamma = (const float*)d_in[4];
    const float* beta  = (const float*)d_in[5];
    const float* Wm1   = (const float*)d_in[6];
    const float* bm1   = (const float*)d_in[7];
    const float* Wm2   = (const float*)d_in[8];
    const float* bm2   = (const float*)d_in[9];
    const float* mem   = (const float*)d_in[10];

    float* out    = (float*)d_out;
    float* logits = out;              // (B, C) first in return order
    float* sdi    = out + B_ * C_;    // (B, D) second

    sdi_pair_kernel<<<B_, 256, 0, stream>>>(X, pairs, W1, b1, gamma, beta, sdi);
    sdi_head_kernel<<<B_ / 16, 32, 0, stream>>>(sdi, Wm1, bm1, Wm2, bm2, mem, logits);
}